// HighGCN_88510686036817
// MI455X (gfx1250) — compile-verified
//
#include <hip/hip_runtime.h>

typedef __attribute__((ext_vector_type(2))) float v2f;
typedef __attribute__((ext_vector_type(8))) float v8f;

#define NN 100000
#define EE 1600000
#define BN_EPS 1e-5f

// ---------------- utility kernels ----------------

__global__ __launch_bounds__(256) void k_zero(float* __restrict__ p, long long n) {
    long long i = (long long)blockIdx.x * 256 + threadIdx.x;
    if (i < n) p[i] = 0.0f;
}

__global__ __launch_bounds__(256) void k_deg(const int* __restrict__ dst, float* __restrict__ deg) {
    int e = blockIdx.x * 256 + threadIdx.x;
    if (e < EE) atomicAdd(&deg[dst[e]], 1.0f);
}

__global__ __launch_bounds__(256) void k_dinv(float* __restrict__ d) {
    int n = blockIdx.x * 256 + threadIdx.x;
    if (n < NN) d[n] = rsqrtf(fmaxf(d[n], 1.0f));
}

// scatter:  agg[dst] += (h * dinv)[src]   (4 features per thread, float4 loads)
__global__ __launch_bounds__(256) void k_scatter(const float* __restrict__ h,
                                                 const float* __restrict__ dinv,
                                                 const int* __restrict__ src,
                                                 const int* __restrict__ dst,
                                                 float* __restrict__ agg) {
    int gid = blockIdx.x * 256 + threadIdx.x;          // EE * 32 threads total
    if (gid >= EE * 32) return;
    int e  = gid >> 5;
    int d4 = (gid & 31) * 4;
    int s = src[e];
    int t = dst[e];
    float w = dinv[s];
    const float4 v = *(const float4*)(h + (size_t)s * 128 + d4);
    float* ap = agg + (size_t)t * 128 + d4;
    atomicAdd(ap + 0, v.x * w);
    atomicAdd(ap + 1, v.y * w);
    atomicAdd(ap + 2, v.z * w);
    atomicAdd(ap + 3, v.w * w);
}

// per-column sum / sumsq over z [N,256]
__global__ __launch_bounds__(256) void k_colstats(const float* __restrict__ z,
                                                  float* __restrict__ csum,
                                                  float* __restrict__ csumsq) {
    int j = threadIdx.x;                 // column 0..255
    int row0 = blockIdx.x * 128;
    float s = 0.f, sq = 0.f;
    for (int r = 0; r < 128; ++r) {
        int row = row0 + r;
        if (row < NN) {
            float v = z[(size_t)row * 256 + j];
            s += v; sq += v * v;
        }
    }
    atomicAdd(&csum[j], s);
    atomicAdd(&csumsq[j], sq);
}

// fold batch-norm into affine per-column scale/shift: zn = z*sA + sB
__global__ __launch_bounds__(256) void k_bn(const float* __restrict__ csum,
                                            const float* __restrict__ csumsq,
                                            const float* __restrict__ gamma,
                                            const float* __restrict__ beta,
                                            float* __restrict__ sA, float* __restrict__ sB) {
    int j = threadIdx.x;
    const float invN = 1.0f / (float)NN;
    float mu  = csum[j] * invN;
    float var = csumsq[j] * invN - mu * mu;
    float rstd = rsqrtf(var + BN_EPS);
    float a = rstd * gamma[j];
    sA[j] = a;
    sB[j] = beta[j] - mu * a;
}

// ---------------- fused WMMA GEMM ----------------
// C[N,OUTC] = Aeff @ W^T (+bias),  W is [OUTC,K] row-major.
// AMODE 0: Aeff = A
// AMODE 1: Aeff = A - aux1*dinv   (high-pass correction; aux1=[N,K] agg, aux2=[N] dinv)
// AMODE 2: Aeff = relu(A*aux1[k] + aux2[k])   (BN+ReLU folded; aux1=sA, aux2=sB per-col)
template <int K, int OUTC, int AMODE>
__global__ __launch_bounds__(256) void gemm_wmma(const float* __restrict__ A,
                                                 const float* __restrict__ aux1,
                                                 const float* __restrict__ aux2,
                                                 const float* __restrict__ W,
                                                 const float* __restrict__ bias,
                                                 float* __restrict__ C) {
    constexpr int PITCH = K + 4;               // 64-bank conflict free, keeps 8B alignment
    __shared__ float As[16 * PITCH];

    const int rowBase = blockIdx.x * 16;       // N % 16 == 0
    const int tid = threadIdx.x;

    // cooperative staging of the fused A tile (16 x K) into LDS
    for (int idx = tid; idx < 16 * K; idx += 256) {
        int r = idx / K;
        int k = idx - r * K;
        int row = rowBase + r;
        float v = A[(size_t)row * K + k];
        if constexpr (AMODE == 1) {
            v -= aux1[(size_t)row * K + k] * aux2[row];
        } else if constexpr (AMODE == 2) {
            v = v * aux1[k] + aux2[k];
            v = v > 0.f ? v : 0.f;
        }
        As[r * PITCH + k] = v;
    }
    __syncthreads();

    const int lane = tid & 31;
    const int wave = tid >> 5;
    const int m  = lane & 15;                  // A/B row / col within tile
    const int hi = lane >> 4;                  // low/high half of wave

    for (int ct = wave; ct < OUTC / 16; ct += 8) {
        const int n = ct * 16 + m;             // output column (== W row)
        const float* Wn = W + (size_t)n * K;
        v8f acc = {};
#pragma unroll
        for (int kk = 0; kk < K; kk += 4) {
            const int kb = kk + 2 * hi;        // ISA 16x4 f32 A layout: lanes 0-15 K={0,1}, 16-31 K={2,3}
            v2f a, b;
            a[0] = As[m * PITCH + kb];
            a[1] = As[m * PITCH + kb + 1];
            b[0] = Wn[kb];                     // B[k][n] = W[n][k]
            b[1] = Wn[kb + 1];
            acc = __builtin_amdgcn_wmma_f32_16x16x4_f32(false, a, false, b,
                                                        (short)0, acc, false, false);
        }
        const float bv = bias ? bias[n] : 0.0f;
#pragma unroll
        for (int r = 0; r < 8; ++r) {          // D layout: VGPR r = M (r + 8*hi), N = lane&15
            int row = rowBase + r + 8 * hi;
            C[(size_t)row * OUTC + n] = acc[r] + bv;
        }
    }
}

// ---------------- driver ----------------

extern "C" void kernel_launch(void* const* d_in, const int* in_sizes, int n_in,
                              void* d_out, int out_size, void* d_ws, size_t ws_size,
                              hipStream_t stream) {
    const float* feat  = (const float*)d_in[0];
    const int*   src   = (const int*)  d_in[1];
    const int*   dst   = (const int*)  d_in[2];
    const float* W0    = (const float*)d_in[3];
    const float* W1    = (const float*)d_in[4];
    const float* Wp1   = (const float*)d_in[5];
    const float* bp1   = (const float*)d_in[6];
    const float* gamma = (const float*)d_in[7];
    const float* beta  = (const float*)d_in[8];
    const float* Wp2   = (const float*)d_in[9];
    const float* bp2   = (const float*)d_in[10];
    float* out = (float*)d_out;

    float* ws      = (float*)d_ws;
    float* dinv    = ws;                               // N (holds deg, then rsqrt in place)
    float* h       = dinv + NN;                        // N*128
    float* agg     = h + (size_t)NN * 128;             // N*128
    float* z       = agg + (size_t)NN * 128;           // N*256
    float* csum    = z + (size_t)NN * 256;             // 256
    float* csumsq  = csum + 256;                       // 256
    float* sA      = csumsq + 256;                     // 256
    float* sB      = sA + 256;                         // 256

    const int rowTiles = NN / 16;                      // 6250
    const long long nh = (long long)NN * 128;

    // degree -> d^{-1/2}
    k_zero<<<(NN + 255) / 256, 256, 0, stream>>>(dinv, NN);
    k_deg<<<(EE + 255) / 256, 256, 0, stream>>>(dst, dinv);
    k_dinv<<<(NN + 255) / 256, 256, 0, stream>>>(dinv);

    // h = feat @ W0^T
    gemm_wmma<128, 128, 0><<<rowTiles, 256, 0, stream>>>(feat, nullptr, nullptr, W0, nullptr, h);

    // high-pass conv 1
    k_zero<<<(int)((nh + 255) / 256), 256, 0, stream>>>(agg, nh);
    k_scatter<<<(EE * 32 + 255) / 256, 256, 0, stream>>>(h, dinv, src, dst, agg);
    // h = (h - agg*dinv) @ W1^T   (in place: tile fully staged to LDS before writeback)
    gemm_wmma<128, 128, 1><<<rowTiles, 256, 0, stream>>>(h, agg, dinv, W1, nullptr, h);

    // high-pass conv 2 + MLP layer 1
    k_zero<<<(int)((nh + 255) / 256), 256, 0, stream>>>(agg, nh);
    k_scatter<<<(EE * 32 + 255) / 256, 256, 0, stream>>>(h, dinv, src, dst, agg);
    gemm_wmma<128, 256, 1><<<rowTiles, 256, 0, stream>>>(h, agg, dinv, Wp1, bp1, z);

    // batch-norm statistics -> per-column affine
    k_zero<<<2, 256, 0, stream>>>(csum, 512);
    k_colstats<<<(NN + 127) / 128, 256, 0, stream>>>(z, csum, csumsq);
    k_bn<<<1, 256, 0, stream>>>(csum, csumsq, gamma, beta, sA, sB);

    // out = relu(z*sA + sB) @ Wp2^T + bp2
    gemm_wmma<256, 128, 2><<<rowTiles, 256, 0, stream>>>(z, sA, sB, Wp2, bp2, out);
}